// attention_block_76940044140789
// MI455X (gfx1250) — compile-verified
//
#include <hip/hip_runtime.h>
#include <hip/hip_bf16.h>

// Problem sizes
#define BSZ 64
#define SEQ 64
#define HD  768
#define NH  12
#define AD  64
// LDS row stride in floats: 68 -> row stride 272B (16B aligned for b128 async
// stores) and bank-conflict free (68 mod 64 = 4; lanes 0..15 hit 16 distinct banks,
// hi half-lanes hit the disjoint +2 set).
#define PAD 68

typedef __attribute__((ext_vector_type(2))) float v2f;
typedef __attribute__((ext_vector_type(8))) float v8f;
typedef __attribute__((ext_vector_type(4))) int v4i;

#define WMMA_F32(A, B, C) \
  __builtin_amdgcn_wmma_f32_16x16x4_f32(false, (A), false, (B), (short)0, (C), false, false)

#if __has_builtin(__builtin_amdgcn_global_load_async_to_lds_b128)
#define USE_ASYNC 1
typedef __attribute__((address_space(1))) v4i gv4i;  // global (AS1) int4
typedef __attribute__((address_space(3))) v4i lv4i;  // LDS (AS3) int4
__device__ __forceinline__ void async_b128(const float* gsrc, float* ldst) {
  __builtin_amdgcn_global_load_async_to_lds_b128((gv4i*)(void*)gsrc,
                                                 (lv4i*)(void*)ldst, 0, 0);
}
#if __has_builtin(__builtin_amdgcn_s_wait_asynccnt)
#define WAIT_ASYNC(n) __builtin_amdgcn_s_wait_asynccnt(n)
#else
#define WAIT_ASYNC(n) asm volatile("s_wait_asynccnt %0" ::"i"(n) : "memory")
#endif
#else
#define USE_ASYNC 0
#endif

// ---- WMMA f32 16x16x4 fragment helpers (wave32) ----
// A (16x4): lanes 0-15 hold M=lane, K={0,1}; lanes 16-31 hold M=lane-16, K={2,3}
// B (4x16): lanes 0-15 hold N=lane, K={0,1}; lanes 16-31 hold N=lane-16, K={2,3}
// C/D (16x16): VGPR r = M=r (lanes 0-15, N=lane) / M=8+r (lanes 16-31, N=lane-16)

__device__ __forceinline__ v2f frag_rowk(const float* p, int ld, int row0, int k0) {
  int lane = threadIdx.x & 31;
  int i = lane & 15;
  int kh = (lane >> 4) << 1;
  v2f f;
  f.x = p[(row0 + i) * ld + k0 + kh];
  f.y = p[(row0 + i) * ld + k0 + kh + 1];
  return f;
}

__device__ __forceinline__ v2f frag_kn(const float* p, int ld, int k0, int n0) {
  int lane = threadIdx.x & 31;
  int i = lane & 15;
  int kh = (lane >> 4) << 1;
  v2f f;
  f.x = p[(k0 + kh) * ld + n0 + i];
  f.y = p[(k0 + kh + 1) * ld + n0 + i];
  return f;
}

__device__ __forceinline__ void store_frag(float* p, int ld, int m0, int n0, v8f acc) {
  int lane = threadIdx.x & 31;
  int i = lane & 15;
  int mh = (lane >> 4) << 3;
#pragma unroll
  for (int r = 0; r < 8; ++r) p[(size_t)(m0 + mh + r) * ld + n0 + i] = acc[r];
}

// ---------------- Kernel 1: QKV projection ----------------
// grid.x = B * NH * 12 ; block = 128 (4 waves; wave w -> M-tile w)
__global__ __launch_bounds__(128) void qkv_proj_kernel(const float* __restrict__ x,
                                                       const float* __restrict__ Wp,
                                                       float* __restrict__ Q,
                                                       float* __restrict__ K,
                                                       float* __restrict__ V) {
  __shared__ __align__(16) float xs[SEQ * PAD];
  __shared__ __align__(16) float wls[16 * PAD];
  int wg = blockIdx.x;
  int ft = wg % 12;
  int n  = (wg / 12) % NH;
  int b  = wg / (12 * NH);
  int tid = threadIdx.x;
  int m0 = (tid >> 5) << 4;

  const float* xb = x + (size_t)b * SEQ * HD;
  const float* wp = Wp + ((size_t)n * 192 + (size_t)ft * 16) * HD;

  v8f acc = {};
  for (int k0 = 0; k0 < HD; k0 += 64) {
    __syncthreads();
    for (int i = tid; i < SEQ * 64; i += 128) {
      int r = i >> 6, c = i & 63;
      xs[r * PAD + c] = xb[(size_t)r * HD + k0 + c];
    }
    for (int i = tid; i < 16 * 64; i += 128) {
      int r = i >> 6, c = i & 63;
      wls[r * PAD + c] = wp[(size_t)r * HD + k0 + c];
    }
    __syncthreads();
#pragma unroll
    for (int kk = 0; kk < 64; kk += 4) {
      v2f A = frag_rowk(xs, PAD, m0, kk);
      v2f B = frag_rowk(wls, PAD, 0, kk);   // B[k][f] = W[f][k]
      acc = WMMA_F32(A, B, acc);
    }
  }

  float* out;
  int col;
  if (ft < 4)      { out = Q; col = ft * 16; }
  else if (ft < 8) { out = K; col = (ft - 4) * 16; }
  else             { out = V; col = (ft - 8) * 16; }
  float* dst = out + ((size_t)b * NH + n) * SEQ * AD;
  store_frag(dst, AD, m0, col, acc);
}

// ---------------- Kernel 2: attention (scores + mask + softmax + ctx) ----------------
// grid.x = B * NH ; block = 128 (wave w -> 16-row strip)
__global__ __launch_bounds__(128) void attn_kernel(const float* __restrict__ Q,
                                                   const float* __restrict__ K,
                                                   const float* __restrict__ V,
                                                   float* __restrict__ scores_out,
                                                   float* __restrict__ Z) {
  __shared__ __align__(16) float qs[SEQ * PAD];
  __shared__ __align__(16) float ks[SEQ * PAD];
  __shared__ __align__(16) float vs[SEQ * PAD];
  __shared__ __align__(16) float ps[SEQ * PAD];

  int wg = blockIdx.x;          // b*NH + n
  int tid = threadIdx.x;
  int lane = tid & 31;
  int m0 = (tid >> 5) << 4;
  size_t base = (size_t)wg * SEQ * AD;

  for (int i = tid; i < SEQ * AD; i += 128) {
    int r = i >> 6, c = i & 63;
    qs[r * PAD + c] = Q[base + i];
    ks[r * PAD + c] = K[base + i];
    vs[r * PAD + c] = V[base + i];
  }
  __syncthreads();

  int li = lane & 15;
  int mh = (lane >> 4) << 3;
  float* sc = scores_out + (size_t)wg * SEQ * SEQ;

#pragma unroll
  for (int nt = 0; nt < 4; ++nt) {
    int n0 = nt << 4;
    v8f acc = {};
#pragma unroll
    for (int kk = 0; kk < AD; kk += 4) {
      v2f A = frag_rowk(qs, PAD, m0, kk);
      v2f B = frag_rowk(ks, PAD, n0, kk);  // B[k][n] = K[n][k]
      acc = WMMA_F32(A, B, acc);
    }
#pragma unroll
    for (int r = 0; r < 8; ++r) {
      int m = m0 + mh + r;
      int nn = n0 + li;
      float s = (nn > m) ? -__builtin_inff() : acc[r] * 0.125f;  // 1/sqrt(64)
      ps[m * PAD + nn] = s;
      sc[(size_t)m * SEQ + nn] = s;
    }
  }
  __syncthreads();

  if (tid < SEQ) {
    int row = tid;
    float mx = -__builtin_inff();
    for (int j = 0; j <= row; ++j) mx = fmaxf(mx, ps[row * PAD + j]);
    float sum = 0.f;
    for (int j = 0; j <= row; ++j) {
      float e = __expf(ps[row * PAD + j] - mx);
      ps[row * PAD + j] = e;
      sum += e;
    }
    float inv = 1.f / sum;
    for (int j = 0; j <= row; ++j) ps[row * PAD + j] *= inv;
    for (int j = row + 1; j < SEQ; ++j) ps[row * PAD + j] = 0.f;
  }
  __syncthreads();

  float* z = Z + base;  // z_flat layout: (b*12+n)*4096 + s*64 + d
#pragma unroll
  for (int nt = 0; nt < 4; ++nt) {
    int n0 = nt << 4;
    v8f acc = {};
#pragma unroll
    for (int kk = 0; kk < SEQ; kk += 4) {
      v2f A = frag_rowk(ps, PAD, m0, kk);
      v2f B = frag_kn(vs, PAD, kk, n0);    // B[k][d] = V[k][d]
      acc = WMMA_F32(A, B, acc);
    }
    store_frag(z, AD, m0, n0, acc);
  }
}

// ---------------- Kernel 3: y = z_flat @ W_lin^T + b_lin ----------------
// M=64, N=4096, K=49152 ; 256 WGs x 128 threads (wave w -> M-tile w, 16 N-cols/WG).
// HBM-bound on W_lin (805 MB): double-buffered GLOBAL_LOAD_ASYNC_TO_LDS_B128 streams
// chunk c+1 while WMMAs consume chunk c (ASYNCcnt tracks in-flight chunk).
#if USE_ASYNC
__device__ __forceinline__ void issue_chunk(const float* Z, const float* Wl, int n0,
                                            int k0, float* zbuf, float* wbuf, int tid) {
  const int Ktot = NH * SEQ * AD;
#pragma unroll
  for (int t = 0; t < 8; ++t) {          // 64x64 z block: 1024 float4, 8 per thread
    int j = tid + (t << 7);
    int r = j >> 4, c4 = (j & 15) << 2;
    async_b128(Z + (size_t)r * Ktot + k0 + c4, zbuf + r * PAD + c4);
  }
#pragma unroll
  for (int t = 0; t < 2; ++t) {          // 16x64 W block: 256 float4, 2 per thread
    int j = tid + (t << 7);
    int r = j >> 4, c4 = (j & 15) << 2;
    async_b128(Wl + (size_t)(n0 + r) * Ktot + k0 + c4, wbuf + r * PAD + c4);
  }
}
#endif

__global__ __launch_bounds__(128) void linear_kernel(const float* __restrict__ Z,
                                                     const float* __restrict__ Wl,
                                                     const float* __restrict__ bl,
                                                     float* __restrict__ y) {
  const int Ktot = NH * SEQ * AD;  // 49152
  const int NCH = Ktot / 64;       // 768 chunks
  int n0 = blockIdx.x << 4;
  int tid = threadIdx.x;
  int m0 = (tid >> 5) << 4;
  v8f acc = {};

#if USE_ASYNC
  __shared__ __align__(16) float zs[2][BSZ * PAD];
  __shared__ __align__(16) float wls[2][16 * PAD];
  issue_chunk(Z, Wl, n0, 0, zs[0], wls[0], tid);
  for (int c = 0; c < NCH; ++c) {
    int cur = c & 1;
    if (c + 1 < NCH) {
      issue_chunk(Z, Wl, n0, (c + 1) << 6, zs[cur ^ 1], wls[cur ^ 1], tid);
      WAIT_ASYNC(10);  // 10 loads/chunk in flight for c+1; chunk c fully landed
    } else {
      WAIT_ASYNC(0);
    }
    __syncthreads();   // all waves' chunk-c data visible
    const float* zb = zs[cur];
    const float* wb = wls[cur];
#pragma unroll
    for (int kk = 0; kk < 64; kk += 4) {
      v2f A = frag_rowk(zb, PAD, m0, kk);
      v2f B = frag_rowk(wb, PAD, 0, kk);  // B[k][n] = W_lin[n][k]
      acc = WMMA_F32(A, B, acc);
    }
    __syncthreads();   // everyone done reading buf before it is refilled (c+2)
  }
#else
  __shared__ __align__(16) float zs[BSZ * PAD];
  __shared__ __align__(16) float wls[16 * PAD];
  for (int k0 = 0; k0 < Ktot; k0 += 64) {
    __syncthreads();
    for (int i = tid; i < BSZ * 64; i += 128) {
      int r = i >> 6, c = i & 63;
      zs[r * PAD + c] = Z[(size_t)r * Ktot + k0 + c];
    }
    for (int i = tid; i < 16 * 64; i += 128) {
      int r = i >> 6, c = i & 63;
      wls[r * PAD + c] = Wl[(size_t)(n0 + r) * Ktot + k0 + c];
    }
    __syncthreads();
#pragma unroll
    for (int kk = 0; kk < 64; kk += 4) {
      v2f A = frag_rowk(zs, PAD, m0, kk);
      v2f B = frag_rowk(wls, PAD, 0, kk);
      acc = WMMA_F32(A, B, acc);
    }
  }
#endif

  int lane = tid & 31;
  int li = lane & 15;
  int mh = (lane >> 4) << 3;
  float bias = bl[n0 + li];
#pragma unroll
  for (int r = 0; r < 8; ++r)
    y[(size_t)(m0 + mh + r) * 4096 + n0 + li] = acc[r] + bias;
}

extern "C" void kernel_launch(void* const* d_in, const int* in_sizes, int n_in,
                              void* d_out, int out_size, void* d_ws, size_t ws_size,
                              hipStream_t stream) {
  (void)in_sizes; (void)n_in; (void)out_size; (void)ws_size;
  const float* x  = (const float*)d_in[0];  // (64,64,768)
  const float* Wp = (const float*)d_in[1];  // (12,192,768)
  const float* Wl = (const float*)d_in[2];  // (4096,49152)
  const float* bl = (const float*)d_in[3];  // (4096,)

  float* out    = (float*)d_out;
  float* y      = out;              // 64*1*64*64 = 262144 floats
  float* scores = out + 262144;     // 64*12*64*64 = 3145728 floats

  const size_t TSZ = (size_t)BSZ * NH * SEQ * AD;  // 3,145,728 floats
  float* Q = (float*)d_ws;
  float* K = Q + TSZ;
  float* V = K + TSZ;
  float* Z = V + TSZ;               // z_flat, (64, 49152)

  qkv_proj_kernel<<<BSZ * NH * 12, 128, 0, stream>>>(x, Wp, Q, K, V);
  attn_kernel<<<BSZ * NH, 128, 0, stream>>>(Q, K, V, scores, Z);
  linear_kernel<<<4096 / 16, 128, 0, stream>>>(Z, Wl, bl, y);
}